// AnchorModule_20461224198371
// MI455X (gfx1250) — compile-verified
//
#include <hip/hip_runtime.h>
#include <hip/hip_bf16.h>
#include <math.h>

typedef __attribute__((ext_vector_type(2))) float v2f;
typedef __attribute__((ext_vector_type(8))) float v8f;

#define BBATCH 16
#define LLEN   32
#define BLN    512          // B*L
#define NPTS   1024
#define FEATS  28
#define NSAMP  8
#define SANCH  81           // 9*3*3
#define MROWS  (BLN*SANCH*NSAMP)   // 331776

// ---------------------------------------------------------------------------
// Stage 1: anchors + distances + stable top-8 + gp feature assembly (K padded to 32)
// One block per (b,t) item; xyz cached in LDS; 8 sequential arg-mins (stable).
// ---------------------------------------------------------------------------
__global__ __launch_bounds__(128)
void top8_gp_kernel(const float* __restrict__ x, const float* __restrict__ g_loc,
                    float* __restrict__ gp)
{
    __shared__ float sx[NPTS], sy[NPTS], sz[NPTS];
    __shared__ float dist[NPTS];
    __shared__ float rd[128];
    __shared__ int   ri[128];
    __shared__ int   win[NSAMP];

    const int bl  = blockIdx.x;
    const int tid = threadIdx.x;
    const float* xb = x + (size_t)bl * NPTS * FEATS;

    for (int p = tid; p < NPTS; p += 128) {
        sx[p] = xb[p*FEATS+0]; sy[p] = xb[p*FEATS+1]; sz[p] = xb[p*FEATS+2];
    }
    const float gx = g_loc[bl*2+0];
    const float gy = g_loc[bl*2+1];
    __syncthreads();

    for (int s = 0; s < SANCH; ++s) {
        const float ax = (float)(s % 3)       * 0.3f - 0.3f + gx;
        const float ay = (float)((s/3) % 3)   * 0.3f - 0.3f + gy;
        const float az = (float)(s/9)         * 0.3f - 0.3f;
        const float an2 = ax*ax + ay*ay + az*az;

        for (int p = tid; p < NPTS; p += 128) {
            float px = sx[p], py = sy[p], pz = sz[p];
            dist[p] = an2 + (px*px + py*py + pz*pz) - 2.0f*(ax*px + ay*py + az*pz);
        }
        __syncthreads();

        for (int k = 0; k < NSAMP; ++k) {
            float bd = 3.4e38f; int bi = NPTS;
            for (int p = tid; p < NPTS; p += 128) {
                float d = dist[p];
                if (d < bd || (d == bd && p < bi)) { bd = d; bi = p; }
            }
            rd[tid] = bd; ri[tid] = bi;
            __syncthreads();
            for (int off = 64; off > 0; off >>= 1) {
                if (tid < off) {
                    float d2 = rd[tid+off]; int i2 = ri[tid+off];
                    if (d2 < rd[tid] || (d2 == rd[tid] && i2 < ri[tid])) { rd[tid]=d2; ri[tid]=i2; }
                }
                __syncthreads();
            }
            if (tid == 0) { win[k] = ri[0]; dist[ri[0]] = 3.4e38f; }
            __syncthreads();
        }

        // write 8 rows x 32 cols (col 31 zero-pad for WMMA K)
        const size_t base = ((size_t)(bl*SANCH + s)) * NSAMP * 32;
        for (int w = tid; w < NSAMP*32; w += 128) {
            int l = w >> 5, c = w & 31;
            int pi = win[l];
            float v;
            if (c < 3) {
                v = (c==0) ? ax : ((c==1) ? ay : az);
            } else if (c < 6) {
                float pc = (c==3) ? sx[pi] : ((c==4) ? sy[pi] : sz[pi]);
                float ac = (c==3) ? ax    : ((c==4) ? ay    : az);
                v = pc - ac;
            } else if (c < 31) {
                v = xb[pi*FEATS + 3 + (c-6)];
            } else {
                v = 0.0f;
            }
            gp[base + (size_t)l*32 + c] = v;
        }
        __syncthreads();
    }
}

// ---------------------------------------------------------------------------
// WMMA fp32 GEMM: C[M x N] = A[M x Kp] * W^T, W is [N x K] row-major.
// One wave per 16x16 output tile; W cached (zero-padded) in LDS.
// Uses V_WMMA_F32_16X16X4_F32 (A 16x4, B 4x16, C 16x16).
// ---------------------------------------------------------------------------
__global__ __launch_bounds__(256)
void gemm_wmma_kernel(const float* __restrict__ A, int lda,
                      const float* __restrict__ W,
                      float* __restrict__ C, int ldc,
                      int Mtiles, int Ntiles, int K)
{
    __shared__ float Wl[64*64];
    const int Kp  = (K + 3) & ~3;
    const int tid = threadIdx.x;
    const int N   = Ntiles * 16;

    for (int i = tid; i < N*Kp; i += 256) {
        int n = i / Kp, k = i % Kp;
        Wl[i] = (k < K) ? W[n*K + k] : 0.0f;
    }
    __syncthreads();

    const int wid  = blockIdx.x * 8 + (tid >> 5);
    const int lane = tid & 31;
    const int mt = wid / Ntiles, nt = wid % Ntiles;
    if (mt >= Mtiles) return;

    const int half = lane >> 4, r = lane & 15;
    const float* Ar = A  + (size_t)(mt*16 + r) * lda;
    const float* Wr = Wl + (nt*16 + r) * Kp;

    v8f acc = {};
    for (int k0 = 0; k0 < Kp; k0 += 4) {
        v2f a, b;
        a.x = Ar[k0 + 2*half]; a.y = Ar[k0 + 2*half + 1];
        b.x = Wr[k0 + 2*half]; b.y = Wr[k0 + 2*half + 1];
        acc = __builtin_amdgcn_wmma_f32_16x16x4_f32(false, a, false, b,
                                                    (short)0, acc, false, false);
    }
    float* Cr = C + (size_t)nt*16 + r;
    for (int j = 0; j < 8; ++j)
        Cr[(size_t)(mt*16 + j + 8*half) * ldc] = acc[j];
}

// ---------------------------------------------------------------------------
// BatchNorm helpers (batch statistics): zero, sum/sumsq reduce, normalize+ReLU.
// Per-channel conv/linear biases cancel inside BN and are never applied.
// ---------------------------------------------------------------------------
__global__ void zero_kernel(float* p, int n)
{
    int i = blockIdx.x*256 + threadIdx.x;
    if (i < n) p[i] = 0.0f;
}

__global__ void bnstat_kernel(const float* __restrict__ X, int Mrows, int C,
                              int rowsPerBlock, float* __restrict__ stats)
{
    int c  = threadIdx.x;                  // blockDim.x == C
    int r0 = blockIdx.x * rowsPerBlock;
    int r1 = r0 + rowsPerBlock; if (r1 > Mrows) r1 = Mrows;
    float s = 0.0f, s2 = 0.0f;
    for (int r = r0; r < r1; ++r) {
        float v = X[(size_t)r*C + c];
        s += v; s2 += v*v;
    }
    atomicAdd(stats + c, s);
    atomicAdd(stats + C + c, s2);
}

__global__ void bnrelu_kernel(float* __restrict__ X, int Mrows, int C,
                              const float* __restrict__ stats,
                              const float* __restrict__ g, const float* __restrict__ bt)
{
    int i = blockIdx.x*256 + threadIdx.x;
    int total = Mrows * C;
    if (i >= total) return;
    int c = i % C;
    float inv = 1.0f / (float)Mrows;
    float m   = stats[c] * inv;
    float var = stats[C + c] * inv - m*m;
    float v = (X[i] - m) * rsqrtf(var + 1e-5f) * g[c] + bt[c];
    X[i] = v > 0.0f ? v : 0.0f;
}

// ---------------------------------------------------------------------------
// Attention softmax over the 8 samples + weighted pooling. One wave per row m.
// attn_b is uniform across the softmax axis -> cancels.
// ---------------------------------------------------------------------------
__global__ __launch_bounds__(256)
void attn_kernel(const float* __restrict__ h3, const float* __restrict__ aw,
                 float* __restrict__ attn_out, float* __restrict__ vec)
{
    const int m    = blockIdx.x * 8 + (threadIdx.x >> 5);
    const int lane = threadIdx.x & 31;
    const float* hb = h3 + (size_t)m * NSAMP * 64;
    const float w1 = aw[lane], w2 = aw[lane + 32];

    float lg[NSAMP];
    for (int l = 0; l < NSAMP; ++l) {
        float p = hb[l*64 + lane]*w1 + hb[l*64 + lane + 32]*w2;
        for (int off = 16; off > 0; off >>= 1) p += __shfl_xor(p, off, 32);
        lg[l] = p;
    }
    float mx = lg[0];
    for (int l = 1; l < NSAMP; ++l) mx = fmaxf(mx, lg[l]);
    float sum = 0.0f;
    for (int l = 0; l < NSAMP; ++l) { lg[l] = expf(lg[l] - mx); sum += lg[l]; }
    float isum = 1.0f / sum;

    float v1 = 0.0f, v2 = 0.0f;
    for (int l = 0; l < NSAMP; ++l) {
        float a = lg[l] * isum;
        v1 += hb[l*64 + lane]      * a;
        v2 += hb[l*64 + lane + 32] * a;
    }
    if (lane < NSAMP) attn_out[(size_t)m*NSAMP + lane] = lg[lane]*isum;
    vec[(size_t)m*64 + lane]      = v1;
    vec[(size_t)m*64 + lane + 32] = v2;
}

// ---------------------------------------------------------------------------
// Direct 3D convs (VALID). Shapes are tiny; one thread per output element.
// ---------------------------------------------------------------------------
__global__ void conv1_kernel(const float* __restrict__ vin, const float* __restrict__ w,
                             float* __restrict__ out)  // (512,9,3,3,64) -> (512,7,96)
{
    int idx = blockIdx.x*256 + threadIdx.x;
    if (idx >= 512*7*96) return;
    int o  = idx % 96;
    int dz = (idx / 96) % 7;
    int b  = idx / (96*7);
    const float* vb = vin + (size_t)b * SANCH * 64;
    float acc = 0.0f;
    for (int kd = 0; kd < 3; ++kd)
        for (int kh = 0; kh < 3; ++kh)
            for (int kw = 0; kw < 3; ++kw) {
                const float* vp = vb + ((dz+kd)*9 + kh*3 + kw) * 64;
                const float* wp = w + (size_t)o*64*27 + kd*9 + kh*3 + kw;
                for (int i = 0; i < 64; ++i)
                    acc += vp[i] * wp[(size_t)i*27];
            }
    out[idx] = acc;
}

__global__ void conv2_kernel(const float* __restrict__ vin, const float* __restrict__ w,
                             float* __restrict__ out)  // (512,7,96) -> (512,3,128)
{
    int idx = blockIdx.x*256 + threadIdx.x;
    if (idx >= 512*3*128) return;
    int o  = idx % 128;
    int dz = (idx / 128) % 3;
    int b  = idx / 384;
    float acc = 0.0f;
    for (int kd = 0; kd < 5; ++kd) {
        const float* vp = vin + (size_t)(b*7 + dz + kd) * 96;
        const float* wp = w + (size_t)o*96*5 + kd;
        for (int i = 0; i < 96; ++i)
            acc += vp[i] * wp[(size_t)i*5];
    }
    out[idx] = acc;
}

__global__ void conv3_kernel(const float* __restrict__ vin, const float* __restrict__ w,
                             float* __restrict__ out)  // (512,3,128) -> (512,64)
{
    int idx = blockIdx.x*256 + threadIdx.x;
    if (idx >= 512*64) return;
    int o = idx % 64;
    int b = idx / 64;
    float acc = 0.0f;
    for (int kd = 0; kd < 3; ++kd) {
        const float* vp = vin + (size_t)(b*3 + kd) * 128;
        const float* wp = w + (size_t)o*128*3 + kd;
        for (int i = 0; i < 128; ++i)
            acc += vp[i] * wp[(size_t)i*3];
    }
    out[idx] = acc;
}

// (b,t,c) -> time-major (t,b,c)
__global__ void seq_init_kernel(const float* __restrict__ v3, float* __restrict__ seq)
{
    int i = blockIdx.x*256 + threadIdx.x;
    if (i >= BBATCH*LLEN*64) return;
    int c = i & 63, t = (i >> 6) & 31, b = i >> 11;
    seq[(t*BBATCH + b)*64 + c] = v3[(b*LLEN + t)*64 + c];
}

// ---------------------------------------------------------------------------
// One LSTM layer, single persistent block. h/c/x in LDS, 32 sequential steps.
// ---------------------------------------------------------------------------
__global__ __launch_bounds__(256)
void lstm_kernel(const float* __restrict__ seq_in, float* __restrict__ seq_out,
                 const float* __restrict__ wih, const float* __restrict__ whh,
                 const float* __restrict__ bih, const float* __restrict__ bhh,
                 const float* __restrict__ h0, const float* __restrict__ c0,
                 float* __restrict__ avec, float* __restrict__ hn, float* __restrict__ cn)
{
    __shared__ float hs[1024], cs[1024], hnew[1024], xs[1024];
    const int tid = threadIdx.x;
    for (int i = tid; i < 1024; i += 256) { hs[i] = h0[i]; cs[i] = c0[i]; }
    __syncthreads();

    for (int t = 0; t < LLEN; ++t) {
        for (int i = tid; i < 1024; i += 256) xs[i] = seq_in[t*1024 + i];
        __syncthreads();
        for (int q = 0; q < 4; ++q) {
            int p = q*256 + tid;          // p = b*64 + j
            int b = p >> 6, j = p & 63;
            float gsum[4];
            for (int g = 0; g < 4; ++g) {
                int row = g*64 + j;
                const float* wi = wih + (size_t)row*64;
                const float* wh = whh + (size_t)row*64;
                const float* xv = xs + b*64;
                const float* hv = hs + b*64;
                float acc = bih[row] + bhh[row];
                for (int k = 0; k < 64; ++k)
                    acc += xv[k]*wi[k] + hv[k]*wh[k];
                gsum[g] = acc;
            }
            float ig = 1.0f/(1.0f + expf(-gsum[0]));
            float fg = 1.0f/(1.0f + expf(-gsum[1]));
            float gg = tanhf(gsum[2]);
            float og = 1.0f/(1.0f + expf(-gsum[3]));
            float cnv = fg*cs[p] + ig*gg;
            float hnv = og*tanhf(cnv);
            cs[p]   = cnv;      // only this thread touches cs[p]
            hnew[p] = hnv;      // hs updated after barrier
            seq_out[t*1024 + p] = hnv;
            if (avec) avec[(b*LLEN + t)*64 + j] = hnv;
        }
        __syncthreads();
        for (int i = tid; i < 1024; i += 256) hs[i] = hnew[i];
        __syncthreads();
    }
    for (int i = tid; i < 1024; i += 256) { hn[i] = hs[i]; cn[i] = cs[i]; }
}

// ---------------------------------------------------------------------------
extern "C" void kernel_launch(void* const* d_in, const int* in_sizes, int n_in,
                              void* d_out, int out_size, void* d_ws, size_t ws_size,
                              hipStream_t stream)
{
    (void)in_sizes; (void)n_in; (void)out_size; (void)ws_size;

    const float* x      = (const float*)d_in[0];
    const float* g_loc  = (const float*)d_in[1];
    const float* h0     = (const float*)d_in[2];
    const float* c0     = (const float*)d_in[3];
    const float* pw1    = (const float*)d_in[4];
    const float* pg1    = (const float*)d_in[6];
    const float* pbt1   = (const float*)d_in[7];
    const float* pw2    = (const float*)d_in[8];
    const float* pg2    = (const float*)d_in[10];
    const float* pbt2   = (const float*)d_in[11];
    const float* pw3    = (const float*)d_in[12];
    const float* pg3    = (const float*)d_in[14];
    const float* pbt3   = (const float*)d_in[15];
    const float* attn_w = (const float*)d_in[16];
    const float* vw1    = (const float*)d_in[18];
    const float* vg1    = (const float*)d_in[20];
    const float* vbt1   = (const float*)d_in[21];
    const float* vw2    = (const float*)d_in[22];
    const float* vg2    = (const float*)d_in[24];
    const float* vbt2   = (const float*)d_in[25];
    const float* vw3    = (const float*)d_in[26];
    const float* vg3    = (const float*)d_in[28];
    const float* vbt3   = (const float*)d_in[29];
    const float* lwih   = (const float*)d_in[30];
    const float* lwhh   = (const float*)d_in[31];
    const float* lbih   = (const float*)d_in[32];
    const float* lbhh   = (const float*)d_in[33];

    float* out  = (float*)d_out;
    float* avec = out;              // 16*32*64     = 32768
    float* attn = out + 32768;      // 41472*8      = 331776
    float* hn   = out + 364544;     // 3*16*64      = 3072
    float* cn   = out + 367616;     // 3*16*64      = 3072

    float* ws = (float*)d_ws;
    const size_t BUF = (size_t)MROWS * 64;
    float* bufA  = ws;              // gp / h2 / (later: vec, conv, seq reuse)
    float* bufB  = ws + BUF;        // h1 / h3
    float* stats = ws + 2*BUF;      // 864 floats of sum/sumsq
    float* st1  = stats;            // 2*32
    float* st2  = stats + 64;       // 2*48
    float* st3  = stats + 160;      // 2*64
    float* stc1 = stats + 288;      // 2*96
    float* stc2 = stats + 480;      // 2*128
    float* stc3 = stats + 736;      // 2*64

    // bufA is dead after GEMM3; reuse for the tail of the pipeline.
    float* vecb = bufA;                        // 41472*64 = 2654208
    float* c1b  = bufA + 2654208;              // 512*7*96 = 344064
    float* c2b  = c1b  + 344064;               // 512*3*128 = 196608
    float* c3b  = c2b  + 196608;               // 512*64 = 32768
    float* seqA = c3b  + 32768;                // 32*16*64
    float* seqB = seqA + 32768;

    const int Mt = MROWS / 16;                 // 20736

    // Stage 1: gather -> gp (padded K=32) in bufA
    top8_gp_kernel<<<BLN, 128, 0, stream>>>(x, g_loc, bufA);
    zero_kernel<<<4, 256, 0, stream>>>(stats, 864);

    // MLP layer 1: gp(31) -> 32   (bufA -> bufB)
    gemm_wmma_kernel<<<Mt*2/8, 256, 0, stream>>>(bufA, 32, pw1, bufB, 32, Mt, 2, 31);
    bnstat_kernel<<<1296, 32, 0, stream>>>(bufB, MROWS, 32, 256, st1);
    bnrelu_kernel<<<(MROWS*32 + 255)/256, 256, 0, stream>>>(bufB, MROWS, 32, st1, pg1, pbt1);

    // MLP layer 2: 32 -> 48   (bufB -> bufA)
    gemm_wmma_kernel<<<Mt*3/8, 256, 0, stream>>>(bufB, 32, pw2, bufA, 48, Mt, 3, 32);
    bnstat_kernel<<<1296, 48, 0, stream>>>(bufA, MROWS, 48, 256, st2);
    bnrelu_kernel<<<(MROWS*48 + 255)/256, 256, 0, stream>>>(bufA, MROWS, 48, st2, pg2, pbt2);

    // MLP layer 3: 48 -> 64   (bufA -> bufB)
    gemm_wmma_kernel<<<Mt*4/8, 256, 0, stream>>>(bufA, 48, pw3, bufB, 64, Mt, 4, 48);
    bnstat_kernel<<<1296, 64, 0, stream>>>(bufB, MROWS, 64, 256, st3);
    bnrelu_kernel<<<(MROWS*64 + 255)/256, 256, 0, stream>>>(bufB, MROWS, 64, st3, pg3, pbt3);

    // Attention pool: h3(bufB) -> attn(out), vec(bufA)
    attn_kernel<<<(BLN*SANCH)/8, 256, 0, stream>>>(bufB, attn_w, attn, vecb);

    // Conv stack with BN+ReLU
    conv1_kernel<<<(512*7*96 + 255)/256, 256, 0, stream>>>(vecb, vw1, c1b);
    bnstat_kernel<<<(3584 + 63)/64, 96, 0, stream>>>(c1b, 3584, 96, 64, stc1);
    bnrelu_kernel<<<(344064 + 255)/256, 256, 0, stream>>>(c1b, 3584, 96, stc1, vg1, vbt1);

    conv2_kernel<<<(512*3*128 + 255)/256, 256, 0, stream>>>(c1b, vw2, c2b);
    bnstat_kernel<<<(1536 + 63)/64, 128, 0, stream>>>(c2b, 1536, 128, 64, stc2);
    bnrelu_kernel<<<(196608 + 255)/256, 256, 0, stream>>>(c2b, 1536, 128, stc2, vg2, vbt2);

    conv3_kernel<<<(512*64 + 255)/256, 256, 0, stream>>>(c2b, vw3, c3b);
    bnstat_kernel<<<(512 + 63)/64, 64, 0, stream>>>(c3b, 512, 64, 64, stc3);
    bnrelu_kernel<<<(32768 + 255)/256, 256, 0, stream>>>(c3b, 512, 64, stc3, vg3, vbt3);

    // LSTM (3 layers, sequential)
    seq_init_kernel<<<128, 256, 0, stream>>>(c3b, seqA);
    lstm_kernel<<<1, 256, 0, stream>>>(seqA, seqB, lwih,        lwhh,        lbih,     lbhh,
                                       h0,        c0,        nullptr, hn,        cn);
    lstm_kernel<<<1, 256, 0, stream>>>(seqB, seqA, lwih+16384,  lwhh+16384,  lbih+256, lbhh+256,
                                       h0+1024,   c0+1024,   nullptr, hn+1024,   cn+1024);
    lstm_kernel<<<1, 256, 0, stream>>>(seqA, seqB, lwih+32768,  lwhh+32768,  lbih+512, lbhh+512,
                                       h0+2048,   c0+2048,   avec,    hn+2048,   cn+2048);
}